// TRNetShared_78013785964670
// MI455X (gfx1250) — compile-verified
//
#include <hip/hip_runtime.h>
#include <math.h>

typedef __attribute__((ext_vector_type(2))) float v2f;
typedef __attribute__((ext_vector_type(8))) float v8f;

#define B_  1024
#define N_  1024
#define M_  8
#define D_  64
#define R_  64
#define C_  10

#define LDS_W      32768      /* W_shared: 64*8*64 floats = 128 KB          */
#define LDS_HALF   10240      /* per-batch slice: At + S + phi + reduction  */
#define AT_STRIDE  66         /* pad 2 -> 8B-aligned b64 loads, no bank conflicts */
#define S_STRIDE   66

// ---- kernel 0: Wl[c][i][r] = sum_m W_last[c][i][m][r]  (exact pad-input sum) ----
__global__ void wl_kernel(const float* __restrict__ Wlast, float* __restrict__ Wl) {
    int tid = blockIdx.x * 256 + threadIdx.x;
    if (tid >= C_ * R_ * R_) return;
    int r  = tid & 63;
    int ci = tid >> 6;                       // c*64 + i
    const float* p = Wlast + ((size_t)ci * 8) * 64 + r;
    float s = 0.f;
    #pragma unroll
    for (int m = 0; m < 8; ++m) s += p[m * 64];
    Wl[tid] = s;                             // layout [c][i][r]
}

// ---- kernel 1: phidiv[b][d][m] = sum over 16 pooled features of phi ----
__global__ void phi_kernel(const float* __restrict__ x, float* __restrict__ phidiv) {
    int tid = blockIdx.x * 256 + threadIdx.x;      // tid = b*64 + d
    if (tid >= B_ * D_) return;
    const float* xp = x + (size_t)tid * 16;        // b*1024 + d*16
    float acc[8];
    #pragma unroll
    for (int m = 0; m < 8; ++m) acc[m] = 0.f;
    for (int n = 0; n < 16; ++n) {
        float a = 1.5707963267948966f * xp[n];
        float sn, cs;
        sincosf(a, &sn, &cs);
        float pc[8], ps[8];
        pc[0] = 1.f; ps[0] = 1.f;
        #pragma unroll
        for (int k = 1; k < 8; ++k) { pc[k] = pc[k-1] * cs; ps[k] = ps[k-1] * sn; }
        #pragma unroll
        for (int m = 0; m < 8; ++m) acc[m] = fmaf(pc[7-m], ps[m], acc[m]);
    }
    float* o = phidiv + (size_t)tid * 8;
    #pragma unroll
    for (int m = 0; m < 8; ++m) o[m] = acc[m];
}

// ---- main kernel: 256 threads = 8 waves = 2 batch elements (4 waves each) ----
__global__ __launch_bounds__(256, 1)
void trnet_kernel(const float* __restrict__ Wsh, const float* __restrict__ phidiv,
                  const float* __restrict__ Wl, float* __restrict__ out)
{
    extern __shared__ float smem[];
    float* Wsm = smem;                                   // [i][j][k] 64*8*64
    const int half = threadIdx.x >> 7;                   // which batch in block
    const int t0   = threadIdx.x & 127;
    float* At  = smem + LDS_W + half * LDS_HALF;         // [n=64][66]: At[n][i] = Amat[i][n]
    float* Ssm = At  + 64 * AT_STRIDE;                   // [m=64][66]: state, row-major
    float* phs = Ssm + 64 * S_STRIDE;                    // [d=64][8]
    float* red = phs + 512;                              // [t0=128][10]

    const int b = blockIdx.x * 2 + half;

    // cooperative load: W_shared -> LDS (once, reused 64 steps)
    {
        const float4* src = (const float4*)Wsh;
        float4* dst = (float4*)Wsm;
        for (int q = 0; q < 32; ++q)
            dst[threadIdx.x + 256 * q] = src[threadIdx.x + 256 * q];
    }
    // phi for this batch
    {
        const float* src = phidiv + (size_t)b * (D_ * M_);
        #pragma unroll
        for (int q = 0; q < 4; ++q) phs[t0 + 128 * q] = src[t0 + 128 * q];
    }
    // S = ones
    for (int q = t0; q < 64 * S_STRIDE; q += 128) Ssm[q] = 1.0f;
    __syncthreads();

    const int wave = t0 >> 5;        // strip id: rows 16*wave .. 16*wave+15
    const int lane = t0 & 31;
    const int lmod = lane & 15;
    const int lhi  = lane >> 4;      // 0/1 (lane group)

    for (int d = 0; d < D_; ++d) {
        // ---- build Amat[i][k] = sum_j phi[j]*W[i][j][k], stored transposed in At ----
        float ph[8];
        #pragma unroll
        for (int j = 0; j < 8; ++j) ph[j] = phs[d * 8 + j];
        #pragma unroll 4
        for (int q = 0; q < 32; ++q) {
            int e  = t0 + 128 * q;
            int ko = e & 63;                 // output column (coalesced lane dim)
            int i  = e >> 6;                 // contraction row
            const float* wp = Wsm + i * 512 + ko;
            float v = 0.f;
            #pragma unroll
            for (int j = 0; j < 8; ++j) v = fmaf(ph[j], wp[j * 64], v);
            At[ko * AT_STRIDE + i] = v;
        }
        __syncthreads();

        // ---- S_new(strip 16x64) = S_old(strip) @ Amat(64x64), fp32 WMMA K=4 ----
        const float* srow = Ssm + (16 * wave + lmod) * S_STRIDE;
        v8f c0 = {}, c1 = {}, c2 = {}, c3 = {};
        #pragma unroll
        for (int kb = 0; kb < 16; ++kb) {
            const int koff = kb * 4 + 2 * lhi;
            // A-operand 16x4: lane=M, VGPR g: K = g + 2*(lane>=16)
            v2f a = *(const v2f*)(srow + koff);
            // B-operand 4x16: lane=N, VGPR g: K = g + 2*(lane>=16)  (read from At transposed)
            const float* ab = At + koff;
            v2f b0 = *(const v2f*)(ab + (lmod +  0) * AT_STRIDE);
            v2f b1 = *(const v2f*)(ab + (lmod + 16) * AT_STRIDE);
            v2f b2 = *(const v2f*)(ab + (lmod + 32) * AT_STRIDE);
            v2f b3 = *(const v2f*)(ab + (lmod + 48) * AT_STRIDE);
            c0 = __builtin_amdgcn_wmma_f32_16x16x4_f32(false, a, false, b0, (short)0, c0, false, false);
            c1 = __builtin_amdgcn_wmma_f32_16x16x4_f32(false, a, false, b1, (short)0, c1, false, false);
            c2 = __builtin_amdgcn_wmma_f32_16x16x4_f32(false, a, false, b2, (short)0, c2, false, false);
            c3 = __builtin_amdgcn_wmma_f32_16x16x4_f32(false, a, false, b3, (short)0, c3, false, false);
        }
        // write-back (strip is wave-private; same-wave LDS ops stay in order)
        // C/D layout: VGPR v -> row M = v + 8*(lane>=16), col N = lane%16
        #pragma unroll
        for (int v = 0; v < 8; ++v) {
            float* prow = Ssm + (16 * wave + v + 8 * lhi) * S_STRIDE + lmod;
            prow[0]  = c0[v];
            prow[16] = c1[v];
            prow[32] = c2[v];
            prow[48] = c3[v];
        }
        __syncthreads();   // At fully consumed; next-iter formation may overwrite
    }

    // ---- traces[b,c] = sum_{r,i} S[r][i] * Wl[c][i][r], then log_softmax ----
    float acc[C_];
    #pragma unroll
    for (int c = 0; c < C_; ++c) acc[c] = 0.f;
    for (int q = 0; q < 32; ++q) {
        int e = t0 + 128 * q;
        int r = e & 63;                       // coalesced over Wl's fastest dim
        int i = e >> 6;
        float s = Ssm[r * S_STRIDE + i];
        #pragma unroll
        for (int c = 0; c < C_; ++c)
            acc[c] = fmaf(s, Wl[c * 4096 + i * 64 + r], acc[c]);
    }
    #pragma unroll
    for (int c = 0; c < C_; ++c) red[t0 * C_ + c] = acc[c];
    __syncthreads();
    if (t0 < C_) {
        float s = 0.f;
        for (int t = 0; t < 128; ++t) s += red[t * C_ + t0];
        At[t0] = s;                           // reuse At as trace buffer
    }
    __syncthreads();
    if (t0 < C_) {
        float mx = At[0];
        #pragma unroll
        for (int c = 1; c < C_; ++c) mx = fmaxf(mx, At[c]);
        float se = 0.f;
        #pragma unroll
        for (int c = 0; c < C_; ++c) se += expf(At[c] - mx);
        out[b * C_ + t0] = At[t0] - (logf(se) + mx);
    }
}

extern "C" void kernel_launch(void* const* d_in, const int* in_sizes, int n_in,
                              void* d_out, int out_size, void* d_ws, size_t ws_size,
                              hipStream_t stream) {
    const float* x     = (const float*)d_in[0];   // (B, N)
    const float* Wsh   = (const float*)d_in[1];   // (R, M, R)
    const float* Wlast = (const float*)d_in[2];   // (C, R, M, R)
    float* out = (float*)d_out;                   // (B, C)
    float* ws  = (float*)d_ws;
    float* Wl     = ws;                           // 40960 floats
    float* phidiv = ws + (C_ * R_ * R_);          // 524288 floats

    wl_kernel<<<(C_ * R_ * R_ + 255) / 256, 256, 0, stream>>>(Wlast, Wl);
    phi_kernel<<<(B_ * D_ + 255) / 256, 256, 0, stream>>>(x, phidiv);

    const size_t shbytes = (size_t)(LDS_W + 2 * LDS_HALF) * sizeof(float);  // 208 KB < 320 KB
    hipFuncSetAttribute((const void*)trnet_kernel,
                        hipFuncAttributeMaxDynamicSharedMemorySize, (int)shbytes);
    trnet_kernel<<<B_ / 2, 256, shbytes, stream>>>(Wsh, phidiv, Wl, out);
}